// deform_conv_25941602467879
// MI455X (gfx1250) — compile-verified
//
#include <hip/hip_runtime.h>
#include <hip/hip_bf16.h>

typedef __attribute__((ext_vector_type(2))) float v2f;
typedef __attribute__((ext_vector_type(8))) float v8f;
typedef __attribute__((ext_vector_type(4))) unsigned int v4u;
typedef __attribute__((ext_vector_type(8))) int v8i;
typedef __attribute__((ext_vector_type(4))) int v4i;

#define B_    8
#define C_    64
#define H_    128
#define W_    128
#define HW_   (H_ * W_)
#define COUT1 18           // offset-conv output channels
#define K1    (C_ * 9)     // 576 contraction for offset conv
#define KK_   9
#define KPG   72           // K per group = CING * KK_
#define XPITCH 20          // padded col pitch for staged window (18 -> 20)

// ---------------------------------------------------------------------------
// Kernel 1: offset = PReLU(conv3x3(x, offset_w) + offset_b)
// Block = 2 waves over one (b, h, 16-wide w segment). The 64ch x 3row x 18col
// im2col window is staged once into LDS (zero-padded); A reads are
// unconditional ds_loads, B reads are unconditional (clamped channel)
// contiguous global_load pairs.
// ---------------------------------------------------------------------------
__global__ __launch_bounds__(64) void offset_conv_wmma(
    const float* __restrict__ x, const float* __restrict__ ow,
    const float* __restrict__ ob, const float* __restrict__ pa,
    float* __restrict__ off)
{
  __shared__ float xs[C_ * 3 * XPITCH];   // 15.4 KB

  const int tid   = threadIdx.x;
  const int lane  = tid & 31;
  const int ntile = tid >> 5;             // wave id = N slice

  const int w0 = (blockIdx.x & 7) * 16;   // W/16 = 8 segments per row
  const int h  = (blockIdx.x >> 3) & (H_ - 1);
  const int b  = blockIdx.x >> 10;

  // ---- stage im2col window: rows h-1..h+1, cols w0-1..w0+16, all channels ----
  for (int i = tid; i < C_ * 3 * 18; i += 64) {
    const int ci = i / 54;
    const int r  = i - ci * 54;
    const int ry = r / 18;                // 0..2 -> image row h+ry-1
    const int rx = r - ry * 18;           // 0..17 -> image col w0+rx-1
    const int hy = h + ry - 1;
    const int wx = w0 + rx - 1;
    float v = 0.f;
    if (hy >= 0 && hy < H_ && wx >= 0 && wx < W_)
      v = x[(((size_t)b * C_ + ci) * H_ + hy) * W_ + wx];
    xs[(ci * 3 + ry) * XPITCH + rx] = v;
  }
  __syncthreads();

  // A lane map (16x4 f32): m = lane%16, K = 2*(lane/16)+j
  const int mi    = lane & 15;
  const int kbase = (lane >> 4) << 1;
  // B lane map (4x16 f32): n = lane%16, K = 2*(lane/16)+j
  const int n  = lane & 15;
  const int co = ntile * 16 + n;
  const bool co_ok = (co < COUT1);
  const int  cow   = co_ok ? co : 0;      // clamp -> unconditional valid loads
  const float* owr = ow + (size_t)cow * K1; // ow flat index = co*576 + k

  v8f acc = {};
  for (int k0 = 0; k0 < K1; k0 += 4) {
    const int kp = k0 + kbase;
    v2f a, bv;
#pragma unroll
    for (int j = 0; j < 2; ++j) {
      const int k  = kp + j;
      const int ci = k / 9;
      const int t  = k - ci * 9;
      const int ky = t / 3;               // staged row (h-1 shift baked in)
      const int kx = t - ky * 3;
      a[j]  = xs[(ci * 3 + ky) * XPITCH + mi + kx];
      const float wv = owr[k];            // contiguous pair -> b64
      bv[j] = co_ok ? wv : 0.f;
    }
    acc = __builtin_amdgcn_wmma_f32_16x16x4_f32(
        false, a, false, bv, (short)0, acc, false, false);
  }

  if (co_ok) {
    const float alpha = pa[0];
    const float bias  = ob[co];
    const int m0 = (lane >> 4) << 3;      // D: M = r + 8*(lane/16), N = lane%16
#pragma unroll
    for (int r = 0; r < 8; ++r) {
      float v = acc[r] + bias;
      v = (v > 0.f) ? v : alpha * v;
      off[((size_t)b * COUT1 + co) * HW_ + h * W_ + w0 + m0 + r] = v;
    }
  }
}

// ---------------------------------------------------------------------------
// Kernel 2: deformable grouped conv.
// Block = 4 waves over one (b, h, 16-wide w segment); wave g computes the
// 16-pos x 16-chan tile for group pair g (block-diagonal B, K = 2*72).
// Stage 1 precomputes, per (pos, tap), 4 validity-folded corner weights and
// 4 pre-clamped linear offsets -> the K-loop gathers are branch-free.
// The 18 KB weight tensor is DMA'd into LDS by the Tensor Data Mover.
// ---------------------------------------------------------------------------
__global__ __launch_bounds__(128) void deform_conv_wmma(
    const float* __restrict__ x, const float* __restrict__ off,
    const float* __restrict__ dw, const float* __restrict__ db,
    float* __restrict__ y)
{
  __shared__ float s_cw[16 * KK_][4];     // validity-folded corner weights
  __shared__ int   s_co[16 * KK_][4];     // pre-clamped linear offsets
  __shared__ float s_dw[C_ * KPG];        // 18.4 KB: all deform weights

  const int tid   = threadIdx.x;
  const int lane  = tid & 31;
  const int gpair = tid >> 5;             // wave id = group pair

  const int w0 = (blockIdx.x & 7) * 16;
  const int h  = (blockIdx.x >> 3) & (H_ - 1);
  const int b  = blockIdx.x >> 10;

  // ---- TDM: DMA the full weight tensor (4608 f32) global -> LDS ----
  // D# per CDNA5 ISA ch.8: 1-D tensor, data_size=4B, tile == tensor.
  if (tid < 32) {
    const unsigned lds_base = (unsigned)(uintptr_t)(void*)&s_dw[0];
    const unsigned long long ga = (unsigned long long)(uintptr_t)dw;
    v4u g0;
    g0[0] = 1u;                                   // count=1, is_restore=0
    g0[1] = lds_base;                             // lds_addr (bytes)
    g0[2] = (unsigned)(ga & 0xFFFFFFFFu);         // global_addr[31:0]
    g0[3] = (unsigned)((ga >> 32) & 0x1FFFFFFu)   // global_addr[56:32]
          | (2u << 30);                           // type = 2 ("image")
    v8i g1;
    g1[0] = (2 << 16);                            // wg_mask=0, data_size=4B
    g1[1] = (int)((C_ * KPG & 0xFFFF) << 16);     // tensor_dim0[15:0] @ [63:48]
    g1[2] = (1 << 16);                            // tensor_dim0[31:16]=0, tensor_dim1=1
    g1[3] = (int)((C_ * KPG) << 16);              // tile_dim0 = 4608 @ [127:112]
    g1[4] = 0;                                    // tile_dim1=0 (unused), tile_dim2=0
    g1[5] = C_ * KPG;                             // tensor_dim0_stride[31:0]
    g1[6] = 0;
    g1[7] = 0;
    v4i gz4 = {0, 0, 0, 0};
    v8i gz8 = {0, 0, 0, 0, 0, 0, 0, 0};
    __builtin_amdgcn_tensor_load_to_lds(g0, g1, gz4, gz4, gz8, 0);
    __builtin_amdgcn_s_wait_tensorcnt((short)0);
  }

  // ---- stage bilinear prep: 16 positions x 9 taps ----
  for (int i = tid; i < 16 * KK_; i += 128) {
    const int mi = i / KK_;
    const int t  = i - mi * KK_;
    const int w  = w0 + mi;
    const int ky = t / 3 - 1;
    const int kx = t - (t / 3) * 3 - 1;
    const size_t obase = ((size_t)b * COUT1 + 2 * t) * HW_ + h * W_ + w;
    const float dy = off[obase];
    const float dx = off[obase + HW_];
    const float py = (float)(h + ky) + dy;
    const float px = (float)(w + kx) + dx;
    const float y0f = floorf(py), x0f = floorf(px);
    const float fy = py - y0f,    fx = px - x0f;
    const int iy = (int)y0f, ix = (int)x0f;
    const float cw[4] = { (1.f - fy) * (1.f - fx), (1.f - fy) * fx,
                          fy * (1.f - fx),         fy * fx };
#pragma unroll
    for (int cn = 0; cn < 4; ++cn) {
      const int yy = iy + (cn >> 1);
      const int xx = ix + (cn & 1);
      const bool valid = (yy >= 0) & (yy < H_) & (xx >= 0) & (xx < W_);
      const int yc = min(max(yy, 0), H_ - 1);
      const int xc = min(max(xx, 0), W_ - 1);
      s_cw[i][cn] = valid ? cw[cn] : 0.f;   // garbage*0 == masked corner
      s_co[i][cn] = yc * W_ + xc;           // always-safe linear offset
    }
  }
  __syncthreads();

  // ---- block-diagonal GEMM over K = 2 groups * 72 ----
  const int mi    = lane & 15;
  const int kbase = (lane >> 4) << 1;
  const int n     = lane & 15;
  const int co    = gpair * 16 + n;       // global output channel
  const int nsub  = n >> 3;               // which group of the pair

  const float* xb = x + (size_t)b * C_ * HW_;

  v8f acc = {};
  for (int k0 = 0; k0 < 2 * KPG; k0 += 4) {
    const int kp   = k0 + kbase;          // even; pair never crosses 72
    const int gsub = kp / KPG;            // 0 or 1 (same for both j)
    const int kk0  = kp - gsub * KPG;
    const bool on  = (gsub == nsub);
    v2f a, bv;
#pragma unroll
    for (int j = 0; j < 2; ++j) {
      const int kk  = kk0 + j;
      const int cig = kk / KK_;
      const int t   = kk - cig * KK_;
      const int c   = (gpair * 2 + gsub) * (C_ / 8) + cig;
      const int idx = mi * KK_ + t;
      const float* xc = xb + (size_t)c * HW_;
      // branch-free bilinear: 2x ds_load_b128 + 4 loads + 4 fma
      float sv = s_cw[idx][0] * xc[s_co[idx][0]];
      sv += s_cw[idx][1] * xc[s_co[idx][1]];
      sv += s_cw[idx][2] * xc[s_co[idx][2]];
      sv += s_cw[idx][3] * xc[s_co[idx][3]];
      a[j] = sv;
      const float wv = s_dw[co * KPG + kk];   // contiguous pair -> b64
      bv[j] = on ? wv : 0.f;
    }
    acc = __builtin_amdgcn_wmma_f32_16x16x4_f32(
        false, a, false, bv, (short)0, acc, false, false);
  }

  const float bias = db[co];
  const int m0 = (lane >> 4) << 3;        // D: M = r + 8*(lane/16), N = lane%16
#pragma unroll
  for (int r = 0; r < 8; ++r) {
    y[((size_t)b * C_ + co) * HW_ + h * W_ + w0 + m0 + r] = acc[r] + bias;
  }
}

extern "C" void kernel_launch(void* const* d_in, const int* in_sizes, int n_in,
                              void* d_out, int out_size, void* d_ws, size_t ws_size,
                              hipStream_t stream) {
  const float* x  = (const float*)d_in[0];
  const float* ow = (const float*)d_in[1];
  const float* ob = (const float*)d_in[2];
  const float* pa = (const float*)d_in[3];
  const float* dw = (const float*)d_in[4];
  const float* db = (const float*)d_in[5];
  float* out = (float*)d_out;
  float* off = (float*)d_ws;              // B*18*H*W floats = 9.4 MB scratch

  const int blocks = B_ * H_ * (W_ / 16); // 8192 row-segment tiles
  offset_conv_wmma<<<blocks, 64, 0, stream>>>(x, ow, ob, pa, off);
  deform_conv_wmma<<<blocks, 128, 0, stream>>>(x, off, dw, db, out);
}